// GKDJSDLoss_37735582663438
// MI455X (gfx1250) — compile-verified
//
#include <hip/hip_runtime.h>
#include <hip/hip_bf16.h>
#include <stdint.h>

// ---------------- tunables / constants ----------------
#define CHUNK 2048           // V-columns per block tile
#define PITCH (CHUNK + 4)    // LDS row pitch in floats: stride%64==4 -> bank-conflict-free
#define ALPHA_C 0.9f
#define INV_T 1.0f           // TEMPERATURE = 1.0
#define LOG_EPS_P (-27.631021115928547f)  // ln(1e-12)

typedef __attribute__((ext_vector_type(2))) float v2f;
typedef __attribute__((ext_vector_type(8))) float v8f;

// ---------------- CDNA5 async-copy helpers (inline asm; portable across toolchains) ----
__device__ __forceinline__ void async_load_b128(unsigned lds_byte_addr, const void* gaddr) {
    // GLOBAL_LOAD_ASYNC_TO_LDS_B128: vdst = LDS byte address VGPR, vaddr = 64-bit global addr
    asm volatile("global_load_async_to_lds_b128 %0, %1, off"
                 :: "v"(lds_byte_addr), "v"(gaddr)
                 : "memory");
}
__device__ __forceinline__ void wait_async0() {
    asm volatile("s_wait_asynccnt 0x0" ::: "memory");
}

// ---------------- kernel 1: per (16-token tile, V-chunk) partial (max, sum exp) --------
extern "C" __global__ void __launch_bounds__(256)
lse_partial_kernel(const float* __restrict__ logits,
                   float* __restrict__ wm, float* __restrict__ wsum,
                   int BT, int V, int NCH) {
    extern __shared__ float buf[];           // 16 x PITCH floats (131,328 B)
    __shared__ float m_sh[16];
    __shared__ float mps[8][16];
    __shared__ float sps[8][17];

    const int tid  = threadIdx.x;
    const int tile = blockIdx.x;
    const int ch   = blockIdx.y;
    const int row0 = tile * 16;
    const int col0 = ch * CHUNK;
    const int clen = min(CHUNK, V - col0);
    const float NEG_INF = -__builtin_inff();

    // Prefill only the invalid region (cols >= valid length) so async writes never race it.
    for (int r = 0; r < 16; ++r) {
        int vlen = (row0 + r < BT) ? clen : 0;
        for (int c = vlen + tid; c < CHUNK; c += 256)
            buf[r * PITCH + c] = NEG_INF;
    }

    // Async DMA of the valid region: 16 rows x clen floats, b128 granules.
    const int c4count = clen >> 2;
    if (c4count > 0) {
        const int total = 16 * c4count;
        for (int j = tid; j < total; j += 256) {
            int r  = j / c4count;
            int c4 = j - r * c4count;
            if (row0 + r < BT) {
                const float* g = logits + (size_t)(row0 + r) * V + col0 + (c4 << 2);
                unsigned l = (unsigned)(size_t)(buf + r * PITCH + (c4 << 2)); // low 32 bits = LDS offset
                async_load_b128(l, (const void*)g);
            }
        }
    }
    // Scalar tail if clen % 4 != 0 (not hit for V=32000, kept for generality).
    const int rem = clen & 3;
    if (rem) {
        const int cb = clen & ~3;
        for (int j = tid; j < 16 * rem; j += 256) {
            int r = j / rem;
            int c = cb + (j - r * rem);
            if (row0 + r < BT)
                buf[r * PITCH + c] = logits[(size_t)(row0 + r) * V + col0 + c];
        }
    }
    wait_async0();
    __syncthreads();

    // Lane mapping matches WMMA f32 16x16x4 A-fragment:
    //   lanes 0-15:  (M=lane,  K=k0+0 / k0+1);  lanes 16-31: (M=lane-16, K=k0+2 / k0+3)
    const int lane   = tid & 31;
    const int wave   = tid >> 5;
    const int row    = lane & 15;
    const int colSel = (lane >> 4) << 1;              // 0 or 2
    const float* rowp = buf + row * PITCH;
    const int kbeg = wave * (CHUNK / 8);              // 8 waves split 2048 cols -> 256 each
    const int kend = kbeg + (CHUNK / 8);

    // ---- phase 1: chunk max per row (LDS-resident, free second pass) ----
    float mloc = NEG_INF;
    for (int k0 = kbeg; k0 < kend; k0 += 4) {
        float2 v = *reinterpret_cast<const float2*>(rowp + k0 + colSel);
        mloc = fmaxf(mloc, fmaxf(v.x, v.y));
    }
    mloc = fmaxf(mloc, __shfl_xor(mloc, 16));         // combine the two half-wave column groups
    if (lane < 16) mps[wave][lane] = mloc;
    __syncthreads();
    if (tid < 16) {
        float mm = NEG_INF;
        for (int w = 0; w < 8; ++w) mm = fmaxf(mm, mps[w][tid]);
        m_sh[tid] = mm;
    }
    __syncthreads();

    // ---- phase 2: sum exp via V_WMMA_F32_16X16X4_F32 with B = ones (f32 row-reduce) ----
    const float mrow = m_sh[row];
    v8f acc = {0.f, 0.f, 0.f, 0.f, 0.f, 0.f, 0.f, 0.f};
    v2f bones; bones.x = 1.0f; bones.y = 1.0f;        // every B element = 1 regardless of layout
    for (int k0 = kbeg; k0 < kend; k0 += 4) {
        float2 v = *reinterpret_cast<const float2*>(rowp + k0 + colSel);
        v2f a;
        a.x = __expf((v.x - mrow) * INV_T);           // v_exp_f32 co-executes with XDL WMMA
        a.y = __expf((v.y - mrow) * INV_T);
        acc = __builtin_amdgcn_wmma_f32_16x16x4_f32(
                  /*neg_a=*/false, a, /*neg_b=*/false, bones,
                  /*c_mod=*/(short)0, acc, /*reuse_a=*/false, /*reuse_b=*/false);
    }
    // D layout: VGPR r, lanes 0-15 -> (M=r, N=lane); lanes 16-31 -> (M=8+r, N=lane-16).
    // Column 0 (lanes 0 and 16) carries the full row sums.
    if (lane == 0 || lane == 16) {
        int rb = (lane >> 4) * 8;
        for (int r = 0; r < 8; ++r) sps[wave][rb + r] = acc[r];
    }
    __syncthreads();
    if (tid < 16) {
        float ss = 0.f;
        for (int w = 0; w < 8; ++w) ss += sps[w][tid];
        size_t o = ((size_t)tile * NCH + ch) * 16 + tid;
        wm[o]   = m_sh[tid];   // raw chunk max
        wsum[o] = ss;          // sum exp((x - m)*invT) over chunk
    }
}

// ---------------- kernel 2: merge partials + gather + JSD, atomic per-batch sums -------
extern "C" __global__ void __launch_bounds__(256)
jsd_kernel(const float* __restrict__ logits, const int* __restrict__ tids,
           const float* __restrict__ tlps, const float* __restrict__ mask,
           const float* __restrict__ wm, const float* __restrict__ wsum,
           float* __restrict__ acc, int BT, int T, int V, int K, int NCH, int B) {
    int i = blockIdx.x * blockDim.x + threadIdx.x;
    if (i >= BT) return;
    const int tile = i >> 4, row = i & 15;
    const float* wmb = wm   + ((size_t)tile * NCH) * 16 + row;
    const float* wsb = wsum + ((size_t)tile * NCH) * 16 + row;

    float M = -__builtin_inff();
    for (int c = 0; c < NCH; ++c) M = fmaxf(M, wmb[(size_t)c * 16]);
    float S = 0.f;
    for (int c = 0; c < NCH; ++c)
        S += wsb[(size_t)c * 16] * __expf((wmb[(size_t)c * 16] - M) * INV_T);
    const float lse = M * INV_T + __logf(S);

    const int Kc = (K > 32) ? 32 : K;
    float pr[32];
    float sumpr = 0.f;
    for (int k = 0; k < Kc; ++k) {
        float p = __expf(tlps[(size_t)i * K + k]);
        pr[k] = p; sumpr += p;
    }
    const float inv_den = 1.0f / fmaxf(sumpr, 1e-8f);

    float jsd = 0.f;
    for (int k = 0; k < Kc; ++k) {
        float p   = pr[k] * inv_den;                         // teacher prob (renormalized)
        int   id  = tids[(size_t)i * K + k];
        float slp = logits[(size_t)i * V + id] * INV_T - lse; // student log-prob
        float q   = __expf(slp);
        float m   = ALPHA_C * p + (1.0f - ALPHA_C) * q;
        float logm = __logf(fmaxf(m, 1e-12f));
        float logp = __logf(fmaxf(p, 1e-12f));
        float logq = fmaxf(slp, LOG_EPS_P);                   // log(max(q,1e-12))
        jsd += ALPHA_C * p * (logp - logm) + (1.0f - ALPHA_C) * q * (logq - logm);
    }
    const float w = mask[i];
    const int b = i / T;
    atomicAdd(&acc[b],     jsd * w);
    atomicAdd(&acc[B + b], w);
}

// ---------------- small helpers ----------------
extern "C" __global__ void init_kernel(float* acc, int n) {
    int i = blockIdx.x * blockDim.x + threadIdx.x;
    if (i < n) acc[i] = 0.f;
}
extern "C" __global__ void final_kernel(const float* __restrict__ acc,
                                        float* __restrict__ out, int B) {
    float s = 0.f;
    for (int b = 0; b < B; ++b) s += acc[b] / fmaxf(acc[B + b], 1.0f);
    out[0] = s / (float)B;
}

// ---------------- host launcher ----------------
extern "C" void kernel_launch(void* const* d_in, const int* in_sizes, int n_in,
                              void* d_out, int out_size, void* d_ws, size_t ws_size,
                              hipStream_t stream) {
    const float* logits = (const float*)d_in[0];
    // d_in[1] (completion_ids) and d_in[5] (prompt_lens) are unused by the reference math.
    const int*   tids   = (const int*)d_in[2];
    const float* tlps   = (const float*)d_in[3];
    const float* mask   = (const float*)d_in[4];

    const int B  = in_sizes[5];
    const int BT = in_sizes[1];
    const int T  = BT / B;
    const int V  = in_sizes[0] / BT;
    const int K  = in_sizes[3] / BT;
    const int NCH    = (V + CHUNK - 1) / CHUNK;
    const int ntiles = (BT + 15) / 16;

    float* wm   = (float*)d_ws;
    float* wsum = wm   + (size_t)ntiles * NCH * 16;
    float* acc  = wsum + (size_t)ntiles * NCH * 16;   // 2*B floats

    const size_t smem = (size_t)16 * PITCH * sizeof(float);  // 131,328 B of WGP LDS
    (void)hipFuncSetAttribute((const void*)lse_partial_kernel,
                              hipFuncAttributeMaxDynamicSharedMemorySize, (int)smem);

    lse_partial_kernel<<<dim3(ntiles, NCH), 256, smem, stream>>>(logits, wm, wsum, BT, V, NCH);
    init_kernel<<<1, 64, 0, stream>>>(acc, 2 * B);
    jsd_kernel<<<(BT + 255) / 256, 256, 0, stream>>>(logits, tids, tlps, mask,
                                                     wm, wsum, acc, BT, T, V, K, NCH, B);
    final_kernel<<<1, 1, 0, stream>>>(acc, (float*)d_out, B);
}